// CTCLayer_5772436046119
// MI455X (gfx1250) — compile-verified
//
#include <hip/hip_runtime.h>

// CTC forward loss on gfx1250 (MI455X).
// B=256 rows, T=1024 steps, C=128 classes, L=64 labels -> S=129 states.
// One block per row (256 blocks x 5 wave32s). The kernel is latency-bound on
// the T=1024 sequential alpha recursion (~134 MB of y_pred is only ~5.8 us at
// 23.3 TB/s), so each step is: 1 workgroup barrier + 3 LDS reads + 1 LDS
// write + 5 TRANS ops. Wave 0 keeps 15 rows of y_pred in flight via CDNA5
// async global->LDS DMA (ASYNCcnt) into a 16-slot power-of-two LDS ring, so
// the 512 B/step row fetch never sits on the dependency chain.

#define CTC_EPS 1e-7f
#define CTC_NEG -1e30f

constexpr int Tc = 1024;
constexpr int Cc = 128;
constexpr int Lc = 64;
constexpr int Sc = 2 * Lc + 1;     // 129 extended states
constexpr int BLANK = Cc - 1;      // 127
constexpr int DEPTH = 15;          // rows in flight (steady state)
constexpr int BUFS  = 16;          // power-of-two ring: slot = t & 15
constexpr int SMASK = BUFS - 1;
constexpr int NTHREADS = 160;      // 5 wave32s; states 0..128 on threads 0..128

typedef int v4i __attribute__((vector_size(16)));

// ---- CDNA5 async global->LDS copy (tracked with ASYNCcnt) -------------------
__device__ __forceinline__ void async_row_b128(const float* g, float* l) {
#if __has_builtin(__builtin_amdgcn_global_load_async_to_lds_b128)
  __builtin_amdgcn_global_load_async_to_lds_b128(
      (__attribute__((address_space(1))) v4i*)g,
      (__attribute__((address_space(3))) v4i*)l,
      /*imm offset*/ 0, /*cpol*/ 0);
#else
  unsigned lds_off =
      (unsigned)(__UINTPTR_TYPE__)(__attribute__((address_space(3))) void*)l;
  asm volatile("global_load_async_to_lds_b128 %0, %1, off"
               :: "v"(lds_off), "v"((unsigned long long)(__UINTPTR_TYPE__)g)
               : "memory");
#endif
}

template <int N>
__device__ __forceinline__ void wait_asynccnt() {
#if __has_builtin(__builtin_amdgcn_s_wait_asynccnt)
  __builtin_amdgcn_s_wait_asynccnt(N);
#else
  asm volatile("s_wait_asynccnt %0" :: "i"(N) : "memory");
#endif
}

// logaddexp3 with max-shift; CTC_NEG sentinel behaves like the reference
// (exp(0) terms collapse, -1e30 + log(k) stays -1e30 in fp32).
__device__ __forceinline__ float logaddexp3(float a1, float a2, float a3) {
  float m = fmaxf(a1, fmaxf(a2, a3));
  float s = __expf(a1 - m) + __expf(a2 - m) + __expf(a3 - m);
  return m + __logf(s);
}

__global__ __launch_bounds__(NTHREADS)
void ctc_fwd_kernel(const int* __restrict__ y_true,
                    const float* __restrict__ y_pred,
                    float* __restrict__ out) {
  __shared__ float rows[BUFS][Cc];       // 16 * 512 B prob-row ring (8 KB)
  __shared__ float abuf[2][Sc + 2];      // double-buffered alpha, 2-elem NEG pad

  const int b   = blockIdx.x;
  const int tid = threadIdx.x;
  const float* gp = y_pred + (size_t)b * Tc * Cc;

  // permanent NEG pad so alpha[s-1]/alpha[s-2] need no bounds checks
  if (tid < 2) { abuf[0][tid] = CTC_NEG; abuf[1][tid] = CTC_NEG; }

  // loop-invariant per-state constants
  const int  s      = tid;
  const bool active = (s < Sc);
  int  ext   = BLANK;
  bool allow = false;
  if (active && (s & 1)) {
    ext = y_true[b * Lc + (s >> 1)];                  // label for odd state
    if (s >= 3) allow = (ext != y_true[b * Lc + ((s - 3) >> 1)]);
  }

  // ---- prologue: wave 0 DMAs rows 0..DEPTH-1 into the ring -----------------
  if (tid < 32) {
    for (int t = 0; t < DEPTH; ++t)
      async_row_b128(gp + t * Cc + 4 * tid, &rows[t][4 * tid]);
  }

  // ---- t = 0 ---------------------------------------------------------------
  wait_asynccnt<DEPTH - 1>();  // wave0: row 0 landed (async loads complete in order)
  __syncthreads();             // publish row 0 + pads to all 5 waves
  int   p = 0;
  float my_alpha = CTC_NEG;
  if (active) {
    float lp0 = __logf(rows[0][ext] + CTC_EPS);
    my_alpha = (s <= 1) ? lp0 : CTC_NEG;
    abuf[p][2 + s] = my_alpha;
  }
  if (tid < 32)                // refill: row DEPTH -> free slot (BUFS > DEPTH)
    async_row_b128(gp + DEPTH * Cc + 4 * tid, &rows[DEPTH & SMASK][4 * tid]);

  // ---- steady state: exactly DEPTH rows in flight, one barrier per step ----
  constexpr int TMAIN = Tc - DEPTH;  // last prefetch target is row Tc-1
  for (int t = 1; t < TMAIN; ++t) {
    wait_asynccnt<DEPTH - 1>();  // wave0: row t landed
    __syncthreads();             // row t ready everywhere; alpha(t-1) visible;
                                 // slot (t-1)&15 free for rewrite (its readers
                                 // consumed it before arriving here)
    float na = CTC_NEG;
    if (active) {
      float a2 = abuf[p][1 + s];                 // alpha[s-1]
      float a3 = allow ? abuf[p][s] : CTC_NEG;   // alpha[s-2]
      float lp = __logf(rows[t & SMASK][ext] + CTC_EPS);
      na = logaddexp3(my_alpha, a2, a3) + lp;
      abuf[p ^ 1][2 + s] = na;
    }
    my_alpha = na;
    p ^= 1;
    if (tid < 32)
      async_row_b128(gp + (t + DEPTH) * Cc + 4 * tid,
                     &rows[(t + DEPTH) & SMASK][4 * tid]);
  }

  // ---- drain: rows TMAIN..Tc-1 all issued; wait for everything once --------
  wait_asynccnt<0>();            // wave0 drains; others pass trivially
  for (int t = TMAIN; t < Tc; ++t) {
    __syncthreads();             // barrier also publishes wave0's drained data
    float na = CTC_NEG;
    if (active) {
      float a2 = abuf[p][1 + s];
      float a3 = allow ? abuf[p][s] : CTC_NEG;
      float lp = __logf(rows[t & SMASK][ext] + CTC_EPS);
      na = logaddexp3(my_alpha, a2, a3) + lp;
      abuf[p ^ 1][2 + s] = na;
    }
    my_alpha = na;
    p ^= 1;
  }

  // ---- epilogue: loss = -logaddexp(alpha[S-1], alpha[S-2]) -----------------
  __syncthreads();
  if (tid == 0) {
    float aN1 = abuf[p][2 + Sc - 1];
    float aN2 = abuf[p][2 + Sc - 2];
    float m   = fmaxf(aN1, aN2);
    out[b] = -(m + __logf(__expf(aN1 - m) + __expf(aN2 - m)));
  }
}

extern "C" void kernel_launch(void* const* d_in, const int* in_sizes, int n_in,
                              void* d_out, int out_size, void* d_ws, size_t ws_size,
                              hipStream_t stream) {
  const int*   y_true = (const int*)d_in[0];    // [B, L] int32
  const float* y_pred = (const float*)d_in[1];  // [B, T, C] fp32
  float*       out    = (float*)d_out;          // [B, 1] fp32
  const int nblocks = out_size;                 // == B == 256
  ctc_fwd_kernel<<<dim3(nblocks), dim3(NTHREADS), 0, stream>>>(y_true, y_pred, out);
}